// GenTargets_77610059039161
// MI455X (gfx1250) — compile-verified
//
#include <hip/hip_runtime.h>
#include <hip/hip_bf16.h>

#ifndef __has_builtin
#define __has_builtin(x) 0
#endif

// ---- problem constants (match the reference) ----
#define NHW    17064      // 12800 + 3200 + 800 + 208 + 56
#define NBOX   64
#define NKP    17
#define BLK    256
#define BIG_AREA 99999999.0f

// LDS staging layout (dwords): boxes | classes | keypoints
#define SH_BOX 0          // 64*4   = 256 dwords
#define SH_CLS 256        // 64     = 64 dwords
#define SH_KP  320        // 64*51  = 3264 dwords
#define SH_DW  3584

#if defined(__gfx1250__) && \
    __has_builtin(__builtin_amdgcn_global_load_async_to_lds_b32) && \
    __has_builtin(__builtin_amdgcn_s_wait_asynccnt)
#define USE_ASYNC_LDS 1
#else
#define USE_ASYNC_LDS 0
#endif

typedef __attribute__((address_space(1))) int as1_int;  // global
typedef __attribute__((address_space(3))) int as3_int;  // LDS

__global__ __launch_bounds__(BLK) void gen_targets_kernel(
    const float* __restrict__ gt_boxes,   // [B,64,4]
    const int*   __restrict__ classes,    // [B,64]
    const float* __restrict__ keypoints,  // [B,64,51]
    float* __restrict__ out, int B)
{
    __shared__ float sh[SH_DW];

    const int b   = blockIdx.y;
    const int tid = threadIdx.x;

    const float* gbox = gt_boxes + (size_t)b * (NBOX * 4);
    const float* gcls = (const float*)(classes + (size_t)b * NBOX); // bitwise stage
    const float* gkp  = keypoints + (size_t)b * (NBOX * 51);

    // gfx1250 speculative prefetch of the GT tables (global_prefetch_b8)
    __builtin_prefetch(gbox, 0, 1);
    __builtin_prefetch(gkp,  0, 1);

    // ---- stage per-batch GT data into LDS ----
#if USE_ASYNC_LDS
    for (int i = tid; i < SH_DW; i += BLK) {
        const float* src = (i < SH_CLS) ? (gbox + i)
                         : (i < SH_KP)  ? (gcls + (i - SH_CLS))
                                        : (gkp  + (i - SH_KP));
        // async copy engine: GLOBAL_LOAD_ASYNC_TO_LDS_B32 (tracked by ASYNCcnt)
        __builtin_amdgcn_global_load_async_to_lds_b32(
            (as1_int*)src,          // global src (addrspace 1)
            (as3_int*)&sh[i],       // LDS dst (addrspace 3, proper aperture cast)
            0, 0);
    }
    __builtin_amdgcn_s_wait_asynccnt(0);
#else
    for (int i = tid; i < SH_DW; i += BLK) {
        float v;
        if (i < SH_CLS)     v = gbox[i];
        else if (i < SH_KP) v = gcls[i - SH_CLS];
        else                v = gkp[i - SH_KP];
        sh[i] = v;
    }
#endif
    __syncthreads();

    const int p = blockIdx.x * BLK + tid;
    if (p >= NHW) return;

    // ---- level lookup: offsets {0,12800,16000,16800,17008}, w = {128,64,32,16,8} ----
    int off, log2w, stridei; float lim0, lim1;
    if (p < 12800)      { off = 0;     log2w = 7; stridei = 8;   lim0 = -1.f;   lim1 = 64.f;     }
    else if (p < 16000) { off = 12800; log2w = 6; stridei = 16;  lim0 = 64.f;   lim1 = 128.f;    }
    else if (p < 16800) { off = 16000; log2w = 5; stridei = 32;  lim0 = 128.f;  lim1 = 256.f;    }
    else if (p < 17008) { off = 16800; log2w = 4; stridei = 64;  lim0 = 256.f;  lim1 = 512.f;    }
    else                { off = 17008; log2w = 3; stridei = 128; lim0 = 512.f;  lim1 = 999999.f; }

    const int   local  = p - off;
    const int   iy     = local >> log2w;
    const int   ix     = local & ((1 << log2w) - 1);
    const float x      = (float)(ix * stridei + (stridei >> 1));
    const float y      = (float)(iy * stridei + (stridei >> 1));
    const float radius = 1.5f * (float)stridei;

    // ---- scan 64 boxes: first-argmin of masked areas ----
    float best  = 3.4e38f;   // > BIG_AREA, so all-negative case picks j=0 (argmin semantics)
    int   bj    = 0;
    bool  anyPos = false;

    #pragma unroll 4
    for (int j = 0; j < NBOX; ++j) {
        const float x1 = sh[SH_BOX + j * 4 + 0];
        const float y1 = sh[SH_BOX + j * 4 + 1];
        const float x2 = sh[SH_BOX + j * 4 + 2];
        const float y2 = sh[SH_BOX + j * 4 + 3];
        const float l  = x  - x1;
        const float t  = y  - y1;
        const float r  = x2 - x;
        const float bo = y2 - y;
        const float omin = fminf(fminf(l, t), fminf(r, bo));
        const float omax = fmaxf(fmaxf(l, t), fmaxf(r, bo));
        const float cx = (x1 + x2) * 0.5f;
        const float cy = (y1 + y2) * 0.5f;
        const float coff = fmaxf(fabsf(x - cx), fabsf(y - cy));
        const bool pos = (omin > 0.f) & (omax > lim0) & (omax <= lim1) & (coff < radius);
        const float area = pos ? (l + r) * (t + bo) : BIG_AREA;
        if (area < best) { best = area; bj = j; }
        anyPos |= pos;
    }

    // ---- gather at best box ----
    const float x1 = sh[SH_BOX + bj * 4 + 0];
    const float y1 = sh[SH_BOX + bj * 4 + 1];
    const float x2 = sh[SH_BOX + bj * 4 + 2];
    const float y2 = sh[SH_BOX + bj * 4 + 3];
    const float l  = x  - x1;
    const float t  = y  - y1;
    const float r  = x2 - x;
    const float bo = y2 - y;
    const float lrmin = fminf(l, r), lrmax = fmaxf(l, r);
    const float tbmin = fminf(t, bo), tbmax = fmaxf(t, bo);
    const float cnt = sqrtf(lrmin * tbmin / (lrmax * tbmax + 1e-10f));
    const int   clsi = __float_as_int(sh[SH_CLS + bj]); // bitwise-staged int32

    const size_t N  = (size_t)B * NHW;
    const size_t gi = (size_t)b * NHW + p;
    float* out_cls = out;
    float* out_cnt = out + N;
    float* out_reg = out + 2 * N;
    float* out_kp  = out + 6 * N;

    out_cls[gi] = anyPos ? (float)clsi : 0.0f;
    out_cnt[gi] = anyPos ? cnt : -1.0f;

    float4 reg4 = anyPos ? make_float4(l, t, r, bo)
                         : make_float4(-1.f, -1.f, -1.f, -1.f);
    *(float4*)(out_reg + gi * 4) = reg4;  // 16B aligned: (2N + 4*gi) % 4 == 0

    float* kpdst = out_kp + gi * 34;      // 8B aligned: 6N and 34*gi are even
    #pragma unroll
    for (int k = 0; k < NKP; ++k) {
        const float vis = sh[SH_KP + bj * 51 + k * 3 + 2];
        float kx = sh[SH_KP + bj * 51 + k * 3 + 0];
        float ky = sh[SH_KP + bj * 51 + k * 3 + 1];
        if (vis == 0.0f) { kx = -99999.0f; ky = -99999.0f; }
        float xs = x - kx;
        float ys = y - ky;
        if (xs > 9999.0f) xs = -1.0f;
        if (ys > 9999.0f) ys = -1.0f;
        if (!anyPos) { xs = -1.0f; ys = -1.0f; }
        *(float2*)(kpdst + 2 * k) = make_float2(xs, ys);
    }
}

extern "C" void kernel_launch(void* const* d_in, const int* in_sizes, int n_in,
                              void* d_out, int out_size, void* d_ws, size_t ws_size,
                              hipStream_t stream) {
    // inputs (setup_inputs order): 0..4 cls_logits (shape-only, unused),
    // 5 gt_boxes [B,64,4] f32, 6 classes [B,64] i32, 7 keypoints [B,64,51] f32
    const float* gt_boxes  = (const float*)d_in[5];
    const int*   classes   = (const int*)d_in[6];
    const float* keypoints = (const float*)d_in[7];
    float*       out       = (float*)d_out;

    const int B = in_sizes[6] / NBOX;  // classes is [B,64]

    dim3 grid((NHW + BLK - 1) / BLK, B);
    gen_targets_kernel<<<grid, dim3(BLK), 0, stream>>>(gt_boxes, classes, keypoints, out, B);
}